// Graph_Sage_GNN_71287867179094
// MI455X (gfx1250) — compile-verified
//
#include <hip/hip_runtime.h>
#include <hip/hip_bf16.h>
#include <math.h>

// ---------------------------------------------------------------------------
// GraphSAGE 2-layer GNN for MI455X (gfx1250, wave32).
//   layer: h = relu( mean_agg(x) @ Wl^T + bl + x @ Wr^T )
// Dense GEMMs use V_WMMA_F32_16X16X4_F32 (fp32 WMMA, one wave per 16x16 tile).
// Edge aggregation uses native global_atomic_add_f32 (bandwidth/L2 bound).
// ---------------------------------------------------------------------------

typedef __attribute__((ext_vector_type(2))) float v2f;
typedef __attribute__((ext_vector_type(8))) float v8f;

#define N_NODES 100000
#define N_EDGES 1600000
#define D_FEAT  128
#define HIDDEN  128
#define OUT_DIM 40

// workspace layout (floats)
#define AGG_OFF 0L                     // N_NODES * 128
#define CNT_OFF 12800000L              // N_NODES
#define H_OFF   12900000L              // N_NODES * 128

__device__ __forceinline__ void atomic_fadd(float* p, float v) {
    __hip_atomic_fetch_add(p, v, __ATOMIC_RELAXED, __HIP_MEMORY_SCOPE_AGENT);
}

// ---------------------------------------------------------------- zero fill
__global__ void fill_zero_kernel(float* __restrict__ p, long n) {
    long i = (long)blockIdx.x * blockDim.x + threadIdx.x;
    long stride = (long)gridDim.x * blockDim.x;
    for (; i < n; i += stride) p[i] = 0.0f;
}

// ------------------------------------------------- edge scatter (mean agg)
// one wave (32 lanes) per edge; each lane moves 4 contiguous floats (float4)
__global__ void scatter_edges_kernel(const float* __restrict__ feat,
                                     const int* __restrict__ src,
                                     const int* __restrict__ dst,
                                     float* __restrict__ agg,
                                     float* __restrict__ cnt,
                                     int add_cnt) {
    long gid = (long)blockIdx.x * blockDim.x + threadIdx.x;
    long e = gid >> 5;
    int lane = (int)(gid & 31);
    if (e >= N_EDGES) return;
    int s = src[e];
    int d = dst[e];
    if (add_cnt && lane == 0) atomic_fadd(&cnt[d], 1.0f);
    const float4* frow = (const float4*)(feat + (long)s * D_FEAT);
    float4 v = frow[lane];
    float* out = agg + (long)d * D_FEAT + lane * 4;
    atomic_fadd(out + 0, v.x);
    atomic_fadd(out + 1, v.y);
    atomic_fadd(out + 2, v.z);
    atomic_fadd(out + 3, v.w);
}

// ------------------------------------------------------------- layer 1 GEMM
// h = relu( (agg/max(cnt,1)) @ Wl1^T + bl1 + x @ Wr1^T )
// grid.x = 6250 row tiles, blockDim = 256 (8 waves) -> wave w = col tile w.
// EXEC stays all-ones (100000 = 6250*16 exactly), required for WMMA.
__global__ void __launch_bounds__(256)
sage_gemm1_kernel(const float* __restrict__ agg,
                  const float* __restrict__ cnt,
                  const float* __restrict__ x,
                  const float* __restrict__ Wl1,
                  const float* __restrict__ bl1,
                  const float* __restrict__ Wr1,
                  float* __restrict__ h) {
    const int lane     = threadIdx.x & 31;
    const int col_tile = threadIdx.x >> 5;      // 0..7
    const int row_tile = blockIdx.x;            // 0..6249
    const int m        = lane & 15;
    const int node     = row_tile * 16 + m;     // A-fragment row for this lane
    const int colN     = col_tile * 16 + m;     // B-fragment column for this lane
    const int khalf    = (lane >> 4) ? 2 : 0;   // ISA 16x4 f32 A layout: K pairs {0,1}/{2,3}

    const float c  = cnt[node];
    const float rc = 1.0f / fmaxf(c, 1.0f);

    const float* arow = agg + (long)node * D_FEAT;
    const float* xrow = x   + (long)node * D_FEAT;
    const float* wlrow = Wl1 + (long)colN * D_FEAT;  // Wl1 is [HIDDEN][D_FEAT]
    const float* wrrow = Wr1 + (long)colN * D_FEAT;

    v8f acc = {};
#pragma unroll
    for (int kb = 0; kb < D_FEAT / 4; ++kb) {
        const int k = kb * 4 + khalf;
        v2f a, b;
        a.x = arow[k]     * rc;
        a.y = arow[k + 1] * rc;
        b.x = wlrow[k];
        b.y = wlrow[k + 1];
        acc = __builtin_amdgcn_wmma_f32_16x16x4_f32(false, a, false, b,
                                                    (short)0, acc, false, false);
        v2f a2, b2;
        a2.x = xrow[k];
        a2.y = xrow[k + 1];
        b2.x = wrrow[k];
        b2.y = wrrow[k + 1];
        acc = __builtin_amdgcn_wmma_f32_16x16x4_f32(false, a2, false, b2,
                                                    (short)0, acc, false, false);
    }

    // C/D layout: VGPR v, lanes 0-15 -> M=v, lanes 16-31 -> M=v+8; N = lane&15
    const float bias = bl1[colN];
    const int mbase = (lane >> 4) ? 8 : 0;
#pragma unroll
    for (int v = 0; v < 8; ++v) {
        const int row = row_tile * 16 + mbase + v;
        h[(long)row * HIDDEN + colN] = fmaxf(acc[v] + bias, 0.0f);
    }
}

// ------------------------------------------------------------- layer 2 GEMM
// out = (agg/max(cnt,1)) @ Wl2^T + bl2 + h @ Wr2^T    (OUT_DIM=40, 3 col tiles)
// blockDim = 96 (3 waves), grid.x = 6250 row tiles.
__global__ void __launch_bounds__(96)
sage_gemm2_kernel(const float* __restrict__ agg,
                  const float* __restrict__ cnt,
                  const float* __restrict__ h,
                  const float* __restrict__ Wl2,
                  const float* __restrict__ bl2,
                  const float* __restrict__ Wr2,
                  float* __restrict__ out) {
    const int lane     = threadIdx.x & 31;
    const int col_tile = threadIdx.x >> 5;      // 0..2
    const int row_tile = blockIdx.x;
    const int m        = lane & 15;
    const int node     = row_tile * 16 + m;
    const int colN     = col_tile * 16 + m;     // may be >= OUT_DIM (masked)
    const int colC     = (colN < OUT_DIM) ? colN : (OUT_DIM - 1);  // clamp loads
    const int khalf    = (lane >> 4) ? 2 : 0;

    const float c  = cnt[node];
    const float rc = 1.0f / fmaxf(c, 1.0f);

    const float* arow  = agg + (long)node * HIDDEN;
    const float* hrow  = h   + (long)node * HIDDEN;
    const float* wlrow = Wl2 + (long)colC * HIDDEN;  // Wl2 is [OUT_DIM][HIDDEN]
    const float* wrrow = Wr2 + (long)colC * HIDDEN;

    v8f acc = {};
#pragma unroll
    for (int kb = 0; kb < HIDDEN / 4; ++kb) {
        const int k = kb * 4 + khalf;
        v2f a, b;
        a.x = arow[k]     * rc;
        a.y = arow[k + 1] * rc;
        b.x = wlrow[k];
        b.y = wlrow[k + 1];
        acc = __builtin_amdgcn_wmma_f32_16x16x4_f32(false, a, false, b,
                                                    (short)0, acc, false, false);
        v2f a2, b2;
        a2.x = hrow[k];
        a2.y = hrow[k + 1];
        b2.x = wrrow[k];
        b2.y = wrrow[k + 1];
        acc = __builtin_amdgcn_wmma_f32_16x16x4_f32(false, a2, false, b2,
                                                    (short)0, acc, false, false);
    }

    const float bias = bl2[colC];
    const int mbase = (lane >> 4) ? 8 : 0;
    if (colN < OUT_DIM) {                      // divergence OK: WMMAs are done
#pragma unroll
        for (int v = 0; v < 8; ++v) {
            const int row = row_tile * 16 + mbase + v;
            out[(long)row * OUT_DIM + colN] = acc[v] + bias;
        }
    }
}

// ---------------------------------------------------------- log_softmax(40)
// one wave per node; lane covers logits {lane, lane+32}
__global__ void logsoftmax_kernel(const float* __restrict__ h2,
                                  float* __restrict__ out) {
    long gid = (long)blockIdx.x * blockDim.x + threadIdx.x;
    long node = gid >> 5;
    int lane = (int)(gid & 31);
    if (node >= N_NODES) return;
    const float* row = h2 + node * OUT_DIM;
    float v0 = row[lane];                                    // lanes 0..31 valid
    float v1 = (lane < OUT_DIM - 32) ? row[lane + 32] : -1e30f;
    float mx = fmaxf(v0, v1);
#pragma unroll
    for (int off = 16; off > 0; off >>= 1)
        mx = fmaxf(mx, __shfl_xor(mx, off, 32));
    float e0 = __expf(v0 - mx);
    float e1 = (lane < OUT_DIM - 32) ? __expf(v1 - mx) : 0.0f;
    float s = e0 + e1;
#pragma unroll
    for (int off = 16; off > 0; off >>= 1)
        s += __shfl_xor(s, off, 32);
    float lse = __logf(s);
    float* orow = out + node * OUT_DIM;
    orow[lane] = v0 - mx - lse;
    if (lane < OUT_DIM - 32) orow[lane + 32] = v1 - mx - lse;
}

// ---------------------------------------------------------------------------
extern "C" void kernel_launch(void* const* d_in, const int* in_sizes, int n_in,
                              void* d_out, int out_size, void* d_ws, size_t ws_size,
                              hipStream_t stream) {
    const float* x    = (const float*)d_in[0];
    const int*   ei   = (const int*)d_in[1];   // [2][N_EDGES]
    const float* Wl1  = (const float*)d_in[2];
    const float* bl1  = (const float*)d_in[3];
    const float* Wr1  = (const float*)d_in[4];
    const float* Wl2  = (const float*)d_in[5];
    const float* bl2  = (const float*)d_in[6];
    const float* Wr2  = (const float*)d_in[7];
    float* out = (float*)d_out;                // [N][40] logits ++ [N][40] log_softmax

    const int* src = ei;
    const int* dst = ei + N_EDGES;

    float* ws  = (float*)d_ws;
    float* agg = ws + AGG_OFF;   // N_NODES*128
    float* cnt = ws + CNT_OFF;   // N_NODES
    float* h   = ws + H_OFF;     // N_NODES*128

    // ---- layer 1 ----
    fill_zero_kernel<<<2048, 256, 0, stream>>>(agg, (long)N_NODES * D_FEAT + N_NODES);
    {
        long threads = (long)N_EDGES * 32;
        int blocks = (int)((threads + 255) / 256);
        scatter_edges_kernel<<<blocks, 256, 0, stream>>>(x, src, dst, agg, cnt, 1);
    }
    sage_gemm1_kernel<<<N_NODES / 16, 256, 0, stream>>>(agg, cnt, x, Wl1, bl1, Wr1, h);

    // ---- layer 2 ----
    fill_zero_kernel<<<2048, 256, 0, stream>>>(agg, (long)N_NODES * D_FEAT);
    {
        long threads = (long)N_EDGES * 32;
        int blocks = (int)((threads + 255) / 256);
        scatter_edges_kernel<<<blocks, 256, 0, stream>>>(h, src, dst, agg, cnt, 0);
    }
    sage_gemm2_kernel<<<N_NODES / 16, 96, 0, stream>>>(agg, cnt, h, Wl2, bl2, Wr2, out);

    // ---- log_softmax ----
    {
        long threads = (long)N_NODES * 32;
        int blocks = (int)((threads + 255) / 256);
        logsoftmax_kernel<<<blocks, 256, 0, stream>>>(out, out + (long)N_NODES * OUT_DIM);
    }
}